// MeshPool_75557064671909
// MI455X (gfx1250) — compile-verified
//
#include <hip/hip_runtime.h>

// MeshPool edge-collapse for MI455X (gfx1250).
// Structure:
//   k_pri      : pri[v] = ||f[v]||^2
//   k_epri     : epri[e] = pri[v0]+pri[v1]
//   k_order    : stable O(E^2) rank sort (keys staged in LDS, 48KB of 320KB)
//   k_init     : copy neighbor matrix into ws (64MB, L2-resident), A = I (64MB)
//   k_collapse : single persistent 1024-thread workgroup; 12288 sequential
//                greedy steps; 4096-wide row/col updates done in parallel
//   k_gemm     : pooled = (A @ F) * alive via V_WMMA_F32_16X16X4_F32,
//                one 16x16 tile per wave32, K=4096 in steps of 4
//   k_tail     : alive (0/1 float) + cnt (float) appended to d_out

#define MP_V 4096
#define MP_D 128
#define MP_E 12288

typedef float v2f __attribute__((ext_vector_type(2)));
typedef float v8f __attribute__((ext_vector_type(8)));

__global__ void k_pri(const float* __restrict__ f, float* __restrict__ pri) {
    int i = blockIdx.x * blockDim.x + threadIdx.x;
    if (i < MP_V) {
        const float* row = f + (size_t)i * MP_D;
        float s = 0.f;
        #pragma unroll 8
        for (int j = 0; j < MP_D; ++j) s += row[j] * row[j];
        pri[i] = s;
    }
}

__global__ void k_epri(const int* __restrict__ edges, const float* __restrict__ pri,
                       float* __restrict__ epri) {
    int e = blockIdx.x * blockDim.x + threadIdx.x;
    if (e < MP_E) epri[e] = pri[edges[e]] + pri[edges[MP_E + e]];
}

// Stable rank sort: rank(i) = #{j : key[j] < key[i]  ||  (key[j]==key[i] && j<i)}
// Reproduces jnp.argsort's stable order. Keys live in LDS (48KB).
__global__ void __launch_bounds__(256) k_order(const float* __restrict__ epri,
                                               int* __restrict__ order) {
    __shared__ float s_key[MP_E];
    for (int j = threadIdx.x; j < MP_E; j += blockDim.x) s_key[j] = epri[j];
    __syncthreads();
    int i = blockIdx.x * blockDim.x + threadIdx.x;
    if (i < MP_E) {
        float key = s_key[i];
        int r = 0;
        for (int j = 0; j < MP_E; ++j) {
            float kj = s_key[j];
            r += (kj < key) | ((kj == key) & (j < i));
        }
        order[r] = i;   // ranks are a permutation: no write conflicts
    }
}

__global__ void k_init(const int* __restrict__ nb_in, int* __restrict__ n,
                       float* __restrict__ A) {
    size_t stride = (size_t)gridDim.x * blockDim.x;
    size_t total  = (size_t)MP_V * MP_V;
    for (size_t i = (size_t)blockIdx.x * blockDim.x + threadIdx.x; i < total; i += stride) {
        n[i] = nb_in[i];
        A[i] = ((i >> 12) == (i & (MP_V - 1))) ? 1.0f : 0.0f;  // V = 4096 = 2^12
    }
}

// Sequential greedy collapse. Single workgroup (32 wave32s on one WGP);
// the 64MB neighbor matrix stays hot in the 192MB L2.
__global__ void __launch_bounds__(1024) k_collapse(
        const int* __restrict__ edges, const int* __restrict__ order,
        int* __restrict__ n, float* __restrict__ A,
        int* __restrict__ alive_out, int* __restrict__ cnt_out) {
    __shared__ int s_alive[MP_V];
    __shared__ int s_v0, s_v1, s_do, s_cnt;
    for (int i = threadIdx.x; i < MP_V; i += blockDim.x) s_alive[i] = 1;
    if (threadIdx.x == 0) s_cnt = MP_V;
    __syncthreads();

    const int target = MP_V / 2;
    for (int it = 0; it < MP_E; ++it) {
        if (threadIdx.x == 0) {
            int e  = order[it];
            int v0 = edges[e];
            int v1 = edges[MP_E + e];
            int d = (s_cnt > target) && s_alive[v0] && s_alive[v1] &&
                    (n[(size_t)v0 * MP_V + v1] == 2);
            s_v0 = v0; s_v1 = v1; s_do = d;
            if (d) { s_alive[v1] = 0; s_cnt = s_cnt - 1; }
        }
        __syncthreads();
        if (s_do) {
            const int v0 = s_v0, v1 = s_v1;
            const size_t r0 = (size_t)v0 * MP_V, r1 = (size_t)v1 * MP_V;
            for (int j = threadIdx.x; j < MP_V; j += blockDim.x) {
                int r = n[r0 + j] + n[r1 + j];
                if (j == v0 || j == v1) r = 0;
                n[r0 + j] = r;                  // row v0
                n[(size_t)j * MP_V + v0] = r;   // col v0 (symmetric)
                n[r1 + j] = 0;                  // row v1
                n[(size_t)j * MP_V + v1] = 0;   // col v1
            }
            for (int j = threadIdx.x; j < MP_V; j += blockDim.x) {
                A[r0 + j] = 0.5f * (A[r0 + j] + A[r1 + j]);
            }
            __threadfence_block();
        }
        __syncthreads();
    }

    for (int i = threadIdx.x; i < MP_V; i += blockDim.x) alive_out[i] = s_alive[i];
    if (threadIdx.x == 0) cnt_out[0] = s_cnt;
}

// pooled = (A @ F) * alive, via V_WMMA_F32_16X16X4_F32.
// One 16x16 output tile per wave32; 8 waves per block; 2048 tiles total.
// ISA layouts (CDNA5 ISA): A 16x4 -> lanes 0-15: M=lane, VGPR0/1 = K0/K1;
// lanes 16-31: K2/K3. B 4x16 -> VGPR0: K0 (lanes 0-15) / K2 (lanes 16-31),
// VGPR1: K1/K3. C/D 16x16 -> VGPR r: M=r (lanes 0-15), M=r+8 (lanes 16-31).
__global__ void __launch_bounds__(256) k_gemm(
        const float* __restrict__ A, const float* __restrict__ F,
        const int* __restrict__ alive, float* __restrict__ out) {
    const int wave = threadIdx.x >> 5;
    const int lane = threadIdx.x & 31;
    const int tile = blockIdx.x * 8 + wave;      // 0..2047
    const int tileM = tile >> 3;                 // 0..255
    const int tileN = tile & 7;                  // 0..7
    const int half = lane >> 4;                  // 0: lanes 0-15, 1: lanes 16-31
    const int l16  = lane & 15;
    const int koff = half * 2;                   // K sub-offset 0 or 2

    const int m  = tileM * 16 + l16;             // A row this lane feeds
    const int nn = tileN * 16 + l16;             // B column this lane feeds
    const float* __restrict__ arow = A + (size_t)m * MP_V + koff;

    v8f c = {};
    for (int k = 0; k < MP_V; k += 4) {
        if ((k & 127) == 0)
            __builtin_prefetch(&arow[k + 256], 0, 0);   // global_prefetch_b8
        v2f a, b;
        a.x = arow[k];
        a.y = arow[k + 1];
        b.x = F[(size_t)(k + koff)     * MP_D + nn];
        b.y = F[(size_t)(k + koff + 1) * MP_D + nn];
        c = __builtin_amdgcn_wmma_f32_16x16x4_f32(
                /*neg_a=*/false, a, /*neg_b=*/false, b,
                /*c_mod=*/(short)0, c, /*reuse_a=*/false, /*reuse_b=*/false);
    }

    const int base_m = tileM * 16 + (half << 3); // +8 for upper half-lanes
    const int col    = tileN * 16 + l16;
    #pragma unroll
    for (int r = 0; r < 8; ++r) {
        const int mr = base_m + r;
        const float mask = alive[mr] ? 1.0f : 0.0f;
        out[(size_t)mr * MP_D + col] = c[r] * mask;
    }
}

__global__ void k_tail(const int* __restrict__ alive, const int* __restrict__ cnt,
                       float* __restrict__ out) {
    int i = blockIdx.x * blockDim.x + threadIdx.x;
    if (i < MP_V) out[(size_t)MP_V * MP_D + i] = alive[i] ? 1.0f : 0.0f;
    if (i == 0)   out[(size_t)MP_V * MP_D + MP_V] = (float)cnt[0];
}

extern "C" void kernel_launch(void* const* d_in, const int* in_sizes, int n_in,
                              void* d_out, int out_size, void* d_ws, size_t ws_size,
                              hipStream_t stream) {
    const float* features = (const float*)d_in[0];   // [V, D] f32
    const int*   edges    = (const int*)d_in[1];     // [2, E] i32
    const int*   neighbor = (const int*)d_in[2];     // [V, V] i32
    float* out = (float*)d_out;                      // pooled | alive | cnt

    // Workspace layout (needs ~128MB + 131KB):
    const size_t NB = (size_t)MP_V * MP_V * sizeof(int);
    char* ws = (char*)d_ws;
    int*   n     = (int*)ws;                         // [V*V] neighbor copy
    float* Amat  = (float*)(ws + NB);                // [V*V] merge matrix
    float* pri   = (float*)(ws + 2 * NB);            // [V]
    float* epri  = pri + MP_V;                       // [E]
    int*   order = (int*)(epri + MP_E);              // [E]
    int*   alive = order + MP_E;                     // [V]
    int*   cnt   = alive + MP_V;                     // [1]

    k_pri <<<(MP_V + 255) / 256, 256, 0, stream>>>(features, pri);
    k_epri<<<(MP_E + 255) / 256, 256, 0, stream>>>(edges, pri, epri);
    k_order<<<(MP_E + 255) / 256, 256, 0, stream>>>(epri, order);
    k_init<<<4096, 256, 0, stream>>>(neighbor, n, Amat);
    k_collapse<<<1, 1024, 0, stream>>>(edges, order, n, Amat, alive, cnt);
    k_gemm<<<256, 256, 0, stream>>>(Amat, features, alive, out);   // 2048 tiles
    k_tail<<<(MP_V + 255) / 256, 256, 0, stream>>>(alive, cnt, out);
}